// Conv2D_86526411145473
// MI455X (gfx1250) — compile-verified
//
#include <hip/hip_runtime.h>
#include <hip/hip_bf16.h>

typedef __attribute__((ext_vector_type(2))) float v2f;
typedef __attribute__((ext_vector_type(8))) float v8f;

#define C_IN_   8
#define C_OUT_  16
#define HW_     1024       // 32*32
#define D0_     3072
#define D1_     8192       // C_IN*HW
#define P_      256        // 16*16 output positions
#define KTAPS   72         // C_IN * 3 * 3
#define KSTEPS  18         // KTAPS / 4 (WMMA f32 K=4)
#define NTILES  192        // D0 / 16
#define NWAVES  8
#define NSLICE  4
#define TILES_PER_SLICE (NTILES / NSLICE)   // 48
#define DSLICE  (D0_ / NSLICE)              // 768

// Load float at base + 32-bit byte offset -> global_load_b32 saddr + voffset form.
__device__ __forceinline__ float ldg_off(const char* __restrict__ base, uint32_t off) {
    return *(const float*)(base + off);
}

// Kernel 1: grid (256 positions, 4 d-slices), 256 threads (8 waves).
// Each block computes, for its slice of d-rows, the concretized partial sums
// for all 16 output channels of position p, and writes 32 floats to ws.
__global__ __launch_bounds__(256) void deeppoly_conv_main(
    const float* __restrict__ l0,  const float* __restrict__ u0,
    const float* __restrict__ lx0, const float* __restrict__ ux0,
    const float* __restrict__ weight,
    float* __restrict__ ws)
{
    __shared__ float s_l0[DSLICE];
    __shared__ float s_u0[DSLICE];
    __shared__ float s_wpos[KTAPS * 16];   // [k][oc], zeroed for padding taps
    __shared__ float s_wneg[KTAPS * 16];
    __shared__ int   s_colb[KTAPS];        // gather column BYTE offset, 0 when padded
    __shared__ float s_pl[NWAVES][32];
    __shared__ float s_pu[NWAVES][32];

    const int tid   = threadIdx.x;
    const int p     = blockIdx.x;
    const int slice = blockIdx.y;
    const int ph = p >> 4, pw = p & 15;
    const int dbase = slice * DSLICE;

    // Stage this slice of the input box (coalesced).
    for (int i = tid; i < DSLICE; i += 256) { s_l0[i] = l0[dbase + i]; s_u0[i] = u0[dbase + i]; }

    // Weights split into pos/neg; padding taps get ZERO weight so the hot loop
    // needs no validity checks (gather offset clamped to 0 instead).
    for (int e = tid; e < KTAPS * 16; e += 256) {
        int k = e >> 4, oc = e & 15;
        int t9 = k % 9;
        int r = ph * 2 - 1 + t9 / 3;
        int c = pw * 2 - 1 + (t9 % 3);
        bool valid = (r >= 0) && (r < 32) && (c >= 0) && (c < 32);
        float w = valid ? weight[oc * KTAPS + k] : 0.f;   // (16,8,3,3) -> oc*72 + k
        s_wpos[k * 16 + oc] = w > 0.f ? w : 0.f;
        s_wneg[k * 16 + oc] = w < 0.f ? w : 0.f;
    }
    if (tid < KTAPS) {
        int ic = tid / 9, t9 = tid % 9;
        int r = ph * 2 - 1 + t9 / 3;
        int c = pw * 2 - 1 + (t9 % 3);
        bool valid = (r >= 0) && (r < 32) && (c >= 0) && (c < 32);
        s_colb[tid] = valid ? ((ic * HW_ + r * 32 + c) << 2) : 0;  // byte offset
    }
    __syncthreads();

    const int wave = tid >> 5;
    const int lane = tid & 31;
    const int m    = lane & 15;            // A row within tile / B col (oc)
    const int koff = (lane >> 4) << 1;     // lanes 16-31 hold K+2, K+3

    // Hoist tile-invariant B fragments and gather byte-offsets into registers.
    v2f bp[KSTEPS], bn[KSTEPS];
    uint32_t c0b[KSTEPS], c1b[KSTEPS];
    #pragma unroll
    for (int s = 0; s < KSTEPS; ++s) {
        const int k0 = s * 4 + koff;
        bp[s]  = (v2f){ s_wpos[k0 * 16 + m], s_wpos[(k0 + 1) * 16 + m] };
        bn[s]  = (v2f){ s_wneg[k0 * 16 + m], s_wneg[(k0 + 1) * 16 + m] };
        c0b[s] = (uint32_t)s_colb[k0];
        c1b[s] = (uint32_t)s_colb[k0 + 1];
    }

    const char* __restrict__ lx0b = (const char*)lx0;   // uniform SGPR bases
    const char* __restrict__ ux0b = (const char*)ux0;

    float acc_l = 0.f, acc_u = 0.f;        // concretized partials, column oc = m

    for (int ti = 0; ti < TILES_PER_SLICE / NWAVES; ++ti) {   // 6 tiles per wave
        const int t   = slice * TILES_PER_SLICE + wave + ti * NWAVES;
        const int d0r = t * 16;
        // Per-lane row byte offset: (d0r+m) * 8192 floats * 4 B = << 15.
        const uint32_t rowb = (uint32_t)(d0r + m) << 15;

        v8f alx = {};   // lx1 tile (16 d-rows x 16 oc)
        v8f aux = {};   // ux1 tile

        #pragma unroll
        for (int s = 0; s < KSTEPS; ++s) {
            const uint32_t o0 = rowb + c0b[s];   // shared voffset for lx0 AND ux0
            const uint32_t o1 = rowb + c1b[s];
            v2f a_l = (v2f){ ldg_off(lx0b, o0), ldg_off(lx0b, o1) };
            v2f a_u = (v2f){ ldg_off(ux0b, o0), ldg_off(ux0b, o1) };
            alx = __builtin_amdgcn_wmma_f32_16x16x4_f32(false, a_l, false, bp[s], (short)0, alx, false, false);
            alx = __builtin_amdgcn_wmma_f32_16x16x4_f32(false, a_u, false, bn[s], (short)0, alx, false, false);
            aux = __builtin_amdgcn_wmma_f32_16x16x4_f32(false, a_u, false, bp[s], (short)0, aux, false, false);
            aux = __builtin_amdgcn_wmma_f32_16x16x4_f32(false, a_l, false, bn[s], (short)0, aux, false, false);
        }

        // Fused concretization. C/D layout: VGPR i -> row M = i + 8*(lane>>4), col N = lane&15.
        const int rloc = (d0r - dbase) + ((lane >> 4) << 3);
        #pragma unroll
        for (int i = 0; i < 8; ++i) {
            const float lv = s_l0[rloc + i], uv = s_u0[rloc + i];
            const float x = alx[i];
            acc_l += (x > 0.f ? lv : uv) * x;       // l0 @ pos(lx1) + u0 @ neg(lx1)
            const float y = aux[i];
            acc_u += (y > 0.f ? uv : lv) * y;       // u0 @ pos(ux1) + l0 @ neg(ux1)
        }
    }

    s_pl[wave][lane] = acc_l;
    s_pu[wave][lane] = acc_u;
    __syncthreads();

    if (tid < 32) {
        const int  oc  = tid & 15;
        const bool isU = tid >= 16;
        float sum = 0.f;
        #pragma unroll
        for (int w = 0; w < NWAVES; ++w)
            sum += isU ? (s_pu[w][oc] + s_pu[w][oc + 16])
                       : (s_pl[w][oc] + s_pl[w][oc + 16]);
        ws[(size_t)(p * NSLICE + slice) * 32 + tid] = sum;
    }
}

// Kernel 2: grid 256 (one per position), 1 wave. Sums the 4 slice partials in
// fixed order (deterministic) and adds the constant-term backsubstitution
// lc1/uc1 = lc0@pl + uc0@nl (+ swap) + bias.
__global__ __launch_bounds__(32) void deeppoly_conv_finish(
    const float* __restrict__ lc0, const float* __restrict__ uc0,
    const float* __restrict__ weight, const float* __restrict__ bias,
    const float* __restrict__ ws, float* __restrict__ out)
{
    const int p   = blockIdx.x;
    const int tid = threadIdx.x;
    const int oc  = tid & 15;
    const bool isU = tid >= 16;
    const int ph = p >> 4, pw = p & 15;

    float sum = 0.f;
    #pragma unroll
    for (int s = 0; s < NSLICE; ++s)
        sum += ws[(size_t)(p * NSLICE + s) * 32 + tid];

    float cacc = bias[oc];                       // lc_c = repeat_interleave(bias, P)
    for (int k = 0; k < KTAPS; ++k) {
        const int ic = k / 9, t9 = k % 9;
        const int r = ph * 2 - 1 + t9 / 3;
        const int c = pw * 2 - 1 + (t9 % 3);
        if (r < 0 || r >= 32 || c < 0 || c >= 32) continue;
        const int j = ic * HW_ + r * 32 + c;
        const float wv = weight[oc * KTAPS + k];
        const float wp = wv > 0.f ? wv : 0.f;
        const float wn = wv < 0.f ? wv : 0.f;
        const float lcv = lc0[j], ucv = uc0[j];
        cacc += isU ? (wp * ucv + wn * lcv)      // uc1
                    : (wp * lcv + wn * ucv);     // lc1
    }
    out[(isU ? 4096 : 0) + oc * P_ + p] = sum + cacc;
}

extern "C" void kernel_launch(void* const* d_in, const int* in_sizes, int n_in,
                              void* d_out, int out_size, void* d_ws, size_t ws_size,
                              hipStream_t stream) {
    const float* l0     = (const float*)d_in[0];
    const float* u0     = (const float*)d_in[1];
    const float* lx0    = (const float*)d_in[2];
    const float* ux0    = (const float*)d_in[3];
    const float* lc0    = (const float*)d_in[4];
    const float* uc0    = (const float*)d_in[5];
    const float* weight = (const float*)d_in[6];
    const float* bias   = (const float*)d_in[7];
    float* ws  = (float*)d_ws;   // needs 256*4*32*4 = 128 KB
    float* out = (float*)d_out;

    deeppoly_conv_main<<<dim3(P_, NSLICE), 256, 0, stream>>>(
        l0, u0, lx0, ux0, weight, ws);
    deeppoly_conv_finish<<<P_, 32, 0, stream>>>(
        lc0, uc0, weight, bias, ws, out);
}